// DTransformer_47785806135769
// MI455X (gfx1250) — compile-verified
//
#include <hip/hip_runtime.h>

// ---------------------------------------------------------------------------
// DTransformer forward for MI455X (gfx1250).  fp32 end-to-end using
// V_WMMA_F32_16X16X4_F32 for every GEMM (projections, QK^T, scores*V).
// Exploits: q==k in b1..b3 (kq_same + same input); in b4 the query rows are
// broadcast of `know` so raw scores are query-row independent (1MB not 537MB)
// and K/V only need the un-broadcast [4,512,256] projections.
// This revision: row-wise decay-softmax is one wave32 per row (16 elements
// per lane, float4 I/O, pure shuffle reductions/scan, ZERO barriers/LDS) --
// ~3x fewer lane-ops on the dominant 262144-row b4 pass.  LayerNorm likewise
// one wave per row.  GEMMs keep 4 N-tiles/wave; AV keeps fused dk halves.
// ---------------------------------------------------------------------------

typedef float v2f __attribute__((ext_vector_type(2)));
typedef float v8f __attribute__((ext_vector_type(8)));

#define NEGV  (-1.0e32f)
#define S_    512
#define D_    256
#define H_    8
#define B_    4
#define NK_   16
#define INV_SQRT_DK 0.17677669529663687f   // 1/sqrt(32)

// ---------------------------------------------------------------------------
// wave32 shuffle reductions
// ---------------------------------------------------------------------------
__device__ __forceinline__ float wred_max(float v) {
  #pragma unroll
  for (int m = 16; m > 0; m >>= 1) v = fmaxf(v, __shfl_xor(v, m, 32));
  return v;
}
__device__ __forceinline__ float wred_sum(float v) {
  #pragma unroll
  for (int m = 16; m > 0; m >>= 1) v += __shfl_xor(v, m, 32);
  return v;
}

// ---------------------------------------------------------------------------
// C[M,256] = A[M,256] @ W[256,256] + bias.  One wave per 16x64 tile
// (4 accumulators -> 4x A-fragment reuse).  grid = (M/16, 4), block = 32.
// ---------------------------------------------------------------------------
__global__ __launch_bounds__(32)
void k_gemm_bias(const float* __restrict__ A, const float* __restrict__ W,
                 const float* __restrict__ bias, float* __restrict__ C) {
  const int tm  = blockIdx.x * 16;
  const int tn0 = blockIdx.y * 64;
  const int lane = threadIdx.x;
  const int lr = lane & 15, hi = lane >> 4;

  const float* arow = A + (size_t)(tm + lr) * D_;
  v8f acc0 = {}, acc1 = {}, acc2 = {}, acc3 = {};
  #pragma unroll 2
  for (int k0 = 0; k0 < D_; k0 += 4) {
    v2f a;
    a.x = arow[k0 + 2 * hi];
    a.y = arow[k0 + 2 * hi + 1];
    const float* w0 = W + (size_t)(k0 + 2 * hi) * D_ + tn0 + lr;
    const float* w1 = W + (size_t)(k0 + 2 * hi + 1) * D_ + tn0 + lr;
    v2f b0, b1, b2, b3;
    b0.x = w0[0];  b0.y = w1[0];
    b1.x = w0[16]; b1.y = w1[16];
    b2.x = w0[32]; b2.y = w1[32];
    b3.x = w0[48]; b3.y = w1[48];
    acc0 = __builtin_amdgcn_wmma_f32_16x16x4_f32(false, a, false, b0, (short)0, acc0, false, false);
    acc1 = __builtin_amdgcn_wmma_f32_16x16x4_f32(false, a, false, b1, (short)0, acc1, false, false);
    acc2 = __builtin_amdgcn_wmma_f32_16x16x4_f32(false, a, false, b2, (short)0, acc2, false, false);
    acc3 = __builtin_amdgcn_wmma_f32_16x16x4_f32(false, a, false, b3, (short)0, acc3, false, false);
  }
  #pragma unroll
  for (int r = 0; r < 8; ++r) {
    float* crow = C + (size_t)(tm + r + 8 * hi) * D_ + tn0 + lr;
    crow[0]  = acc0[r] + bias[tn0 + lr];
    crow[16] = acc1[r] + bias[tn0 + 16 + lr];
    crow[32] = acc2[r] + bias[tn0 + 32 + lr];
    crow[48] = acc3[r] + bias[tn0 + 48 + lr];
  }
}

// ---------------------------------------------------------------------------
// raw[u, m, n] = (1/sqrt(32)) * sum_d Q[b*a_bstride + m*256 + h*32 + d]
//                              * K[b*S*256 + n*256 + h*32 + d]
// u = b*H + h.  One wave per 16x64 tile of keys (4 accumulators).
// grid = (512/64, Mrows/16, B*H), block = 32.
// out layout: ((u*Mrows + m) * 512 + n)
// ---------------------------------------------------------------------------
__global__ __launch_bounds__(32)
void k_qk(const float* __restrict__ Q, const float* __restrict__ K,
          float* __restrict__ OUT, int Mrows, long a_bstride) {
  const int u = blockIdx.z;
  const int b = u / H_, h = u % H_;
  const int tn0 = blockIdx.x * 64;   // key tile base
  const int tm  = blockIdx.y * 16;   // query-row tile
  const int lane = threadIdx.x;
  const int lr = lane & 15, hi = lane >> 4;

  const float* Ab = Q + (size_t)b * a_bstride + h * 32 + (size_t)(tm + lr) * D_;
  const float* Kb = K + (size_t)b * S_ * D_ + h * 32 + (size_t)(tn0 + lr) * D_;

  v8f acc0 = {}, acc1 = {}, acc2 = {}, acc3 = {};
  #pragma unroll
  for (int k0 = 0; k0 < 32; k0 += 4) {
    v2f a;
    a.x = Ab[k0 + 2 * hi];
    a.y = Ab[k0 + 2 * hi + 1];
    v2f b0, b1, b2, b3;                      // B[k][n] = K[n][k]
    b0.x = Kb[k0 + 2 * hi];                 b0.y = Kb[k0 + 2 * hi + 1];
    b1.x = Kb[16 * D_ + k0 + 2 * hi];       b1.y = Kb[16 * D_ + k0 + 2 * hi + 1];
    b2.x = Kb[32 * D_ + k0 + 2 * hi];       b2.y = Kb[32 * D_ + k0 + 2 * hi + 1];
    b3.x = Kb[48 * D_ + k0 + 2 * hi];       b3.y = Kb[48 * D_ + k0 + 2 * hi + 1];
    acc0 = __builtin_amdgcn_wmma_f32_16x16x4_f32(false, a, false, b0, (short)0, acc0, false, false);
    acc1 = __builtin_amdgcn_wmma_f32_16x16x4_f32(false, a, false, b1, (short)0, acc1, false, false);
    acc2 = __builtin_amdgcn_wmma_f32_16x16x4_f32(false, a, false, b2, (short)0, acc2, false, false);
    acc3 = __builtin_amdgcn_wmma_f32_16x16x4_f32(false, a, false, b3, (short)0, acc3, false, false);
  }
  #pragma unroll
  for (int r = 0; r < 8; ++r) {
    float* orow = OUT + ((size_t)u * Mrows + tm + r + 8 * hi) * S_ + tn0 + lr;
    orow[0]  = acc0[r] * INV_SQRT_DK;
    orow[16] = acc1[r] * INV_SQRT_DK;
    orow[32] = acc2[r] * INV_SQRT_DK;
    orow[48] = acc3[r] * INV_SQRT_DK;
  }
}

// ---------------------------------------------------------------------------
// Distance-decay double softmax.  ONE WAVE32 PER ROW, 16 contiguous elements
// per lane, float4 I/O, pure shuffle reductions + local prefix + wave scan.
// No LDS, no barriers.  8 independent waves per 256-thread block.
// b4mode=0:  raw row = RAW[(u*512 + qi)*512], out same layout (in-place OK).
// b4mode=1:  raw row = RAW[(u*16 + j)*512] (query-row independent),
//            out = k_scores layout (((b*H+h)*512 + qi)*16 + j)*512 + ki.
// ---------------------------------------------------------------------------
__global__ __launch_bounds__(256)
void k_rowwise(const float* __restrict__ RAW, float* __restrict__ OUT,
               const float* __restrict__ gammas, int peek, int maxout, int b4mode) {
  const int ln = threadIdx.x & 31;
  const int wv = threadIdx.x >> 5;
  int rid = blockIdx.x * 8 + wv;

  int qi, u;
  size_t rawoff, outoff;
  if (b4mode) {
    qi = rid & 511; rid >>= 9;
    const int j = rid & 15;
    u = rid >> 4;                                     // u = b*H + h
    rawoff = ((size_t)u * 16 + j) * S_;
    outoff = (((size_t)u * S_ + qi) * NK_ + j) * S_;
  } else {
    qi = rid & 511;
    u = rid >> 9;
    rawoff = ((size_t)u * S_ + qi) * S_;
    outoff = rawoff;
  }
  const int h   = u % H_;
  const int base = ln * 16;
  const int lim  = peek ? qi : (qi - 1);              // masked: index <= lim

  // load 16 contiguous elements (4 x b128)
  float e[16];
  {
    const float4* r4 = (const float4*)(RAW + rawoff + base);
    float4 a = r4[0], b = r4[1], c = r4[2], d = r4[3];
    e[0]=a.x; e[1]=a.y; e[2]=a.z;  e[3]=a.w;
    e[4]=b.x; e[5]=b.y; e[6]=b.z;  e[7]=b.w;
    e[8]=c.x; e[9]=c.y; e[10]=c.z; e[11]=c.w;
    e[12]=d.x; e[13]=d.y; e[14]=d.z; e[15]=d.w;
  }

  // ---- first softmax ----
  float lm = NEGV;
  #pragma unroll
  for (int i = 0; i < 16; ++i) {
    const float xi = (base + i <= lim) ? e[i] : NEGV;
    lm = fmaxf(lm, xi);
  }
  const float m1 = wred_max(lm);

  float p[16];                    // exp values -> then prefix of sm
  float lsum = 0.0f;
  #pragma unroll
  for (int i = 0; i < 16; ++i) {
    const float xi = (base + i <= lim) ? e[i] : NEGV;
    const float ex = expf(xi - m1);
    p[i] = ex;
    lsum += ex;
  }
  const float sum1 = wred_sum(lsum);
  const float inv1 = 1.0f / sum1;

  // local inclusive prefix of sm = p*inv1
  float run = 0.0f;
  #pragma unroll
  for (int i = 0; i < 16; ++i) {
    run += p[i] * inv1;
    p[i] = run;
  }
  // wave-level exclusive offset + grand total
  float sc = run;
  #pragma unroll
  for (int d = 1; d < 32; d <<= 1) {
    const float t = __shfl_up(sc, d, 32);
    if (ln >= d) sc += t;
  }
  const float off   = sc - run;
  const float total = __shfl(sc, 31, 32);

  // ---- decay + second softmax ----
  const float g = -fabsf(gammas[h]);
  float lm2 = NEGV;
  #pragma unroll
  for (int i = 0; i < 16; ++i) {
    const float distcum = p[i] + off;
    const float pos  = fabsf((float)(base + i - qi));
    const float dist = sqrtf(fmaxf((total - distcum) * pos, 0.0f));
    float eff = expf(dist * g);
    eff = fminf(fmaxf(eff, 1e-5f), 1e5f);
    const float s2 = e[i] * eff;
    const float x2 = (base + i <= lim) ? s2 : NEGV;
    p[i] = x2;                                   // reuse p[] for x2
    lm2 = fmaxf(lm2, x2);
  }
  const float m2 = wred_max(lm2);

  float ls2 = 0.0f;
  #pragma unroll
  for (int i = 0; i < 16; ++i) {
    const float ex = expf(p[i] - m2);
    e[i] = ex;                                   // reuse e[] for e2
    ls2 += ex;
  }
  const float sum2 = wred_sum(ls2);
  const float inv2 = 1.0f / sum2;

  float lmx = 0.0f;
  #pragma unroll
  for (int i = 0; i < 16; ++i) {
    const float vv = (base + i <= lim) ? e[i] * inv2 : 0.0f;
    e[i] = vv;
    lmx = fmaxf(lmx, vv);
  }
  if (maxout) {
    const float mx = wred_max(lmx);
    const float scale = fminf(1.0f / (mx + 1e-8f), 5.0f);
    #pragma unroll
    for (int i = 0; i < 16; ++i) e[i] *= scale;
  }

  float4* o4 = (float4*)(OUT + outoff + base);
  o4[0] = make_float4(e[0],  e[1],  e[2],  e[3]);
  o4[1] = make_float4(e[4],  e[5],  e[6],  e[7]);
  o4[2] = make_float4(e[8],  e[9],  e[10], e[11]);
  o4[3] = make_float4(e[12], e[13], e[14], e[15]);
}

// ---------------------------------------------------------------------------
// out[bb, qi, h*32 + d] = sum_ki SC[...] * V[(b*512+ki)*256 + h*32 + d]
// Both dk halves per wave (2 accumulators, single SC fragment load).
// u = (b*NKx + j)*H + h.  grid = (512/16, B*NKx*H), block = 32.
// ---------------------------------------------------------------------------
__global__ __launch_bounds__(32)
void k_av(const float* __restrict__ SC, const float* __restrict__ V,
          float* __restrict__ OUT, int NKx) {
  const int u  = blockIdx.y;
  const int h  = u % H_;
  const int j  = (u / H_) % NKx;
  const int b  = u / (H_ * NKx);
  const int tm = blockIdx.x * 16;     // query tile
  const int lane = threadIdx.x;
  const int lr = lane & 15, hi = lane >> 4;

  const size_t rowpitch = (size_t)NKx * S_;
  const float* scb = SC + ((size_t)(b * H_ + h) * S_ + tm + lr) * rowpitch + (size_t)j * S_;
  const float* vb  = V + (size_t)b * S_ * D_ + h * 32;

  v8f acc0 = {}, acc1 = {};
  #pragma unroll 2
  for (int k0 = 0; k0 < S_; k0 += 4) {
    v2f a;
    a.x = scb[k0 + 2 * hi];
    a.y = scb[k0 + 2 * hi + 1];
    const float* v0 = vb + (size_t)(k0 + 2 * hi) * D_;
    const float* v1 = vb + (size_t)(k0 + 2 * hi + 1) * D_;
    v2f b0, b1;
    b0.x = v0[lr];      b0.y = v1[lr];
    b1.x = v0[16 + lr]; b1.y = v1[16 + lr];
    acc0 = __builtin_amdgcn_wmma_f32_16x16x4_f32(false, a, false, b0, (short)0, acc0, false, false);
    acc1 = __builtin_amdgcn_wmma_f32_16x16x4_f32(false, a, false, b1, (short)0, acc1, false, false);
  }
  #pragma unroll
  for (int r = 0; r < 8; ++r) {
    float* orow = OUT + ((size_t)(b * NKx + j) * S_ + tm + r + 8 * hi) * D_ + h * 32;
    orow[lr]      = acc0[r];
    orow[16 + lr] = acc1[r];
  }
}

// ---------------------------------------------------------------------------
// Residual + LayerNorm.  ONE WAVE32 PER ROW (8 elements/lane, no barriers).
// mode=0: OUT[row] = LN(X[row] + DELTA[row]).
// mode=1 (b4): X = know (residual is know[j] broadcast); output scattered
// directly into z layout: (((b*512+s)*16 + j)*256 + d), row = (b*16+j)*512+s.
// ---------------------------------------------------------------------------
__global__ __launch_bounds__(256)
void k_add_ln(const float* __restrict__ X, const float* __restrict__ DELTA,
              const float* __restrict__ g, const float* __restrict__ bta,
              float* __restrict__ OUT, int mode) {
  const int ln = threadIdx.x & 31;
  const int wv = threadIdx.x >> 5;
  const size_t row = (size_t)blockIdx.x * 8 + wv;
  const int base = ln * 8;

  const float* xrow;
  size_t outbase;
  if (mode) {
    const int b = (int)(row >> 13);          // row / (16*512)
    const int j = (int)((row >> 9) & 15);
    const int s = (int)(row & 511);
    xrow = X + (size_t)j * D_;               // know[j]
    outbase = (((size_t)b * S_ + s) * NK_ + j) * D_;
  } else {
    xrow = X + row * D_;
    outbase = row * D_;
  }
  const float* drow = DELTA + row * D_;

  float v[8];
  {
    const float4* x4 = (const float4*)(xrow + base);
    const float4* d4 = (const float4*)(drow + base);
    float4 xa = x4[0], xb = x4[1], da = d4[0], db = d4[1];
    v[0]=xa.x+da.x; v[1]=xa.y+da.y; v[2]=xa.z+da.z; v[3]=xa.w+da.w;
    v[4]=xb.x+db.x; v[5]=xb.y+db.y; v[6]=xb.z+db.z; v[7]=xb.w+db.w;
  }
  float ls = 0.0f;
  #pragma unroll
  for (int i = 0; i < 8; ++i) ls += v[i];
  const float mu = wred_sum(ls) * (1.0f / 256.0f);
  float lv = 0.0f;
  #pragma unroll
  for (int i = 0; i < 8; ++i) { const float d = v[i] - mu; lv += d * d; }
  const float rstd = rsqrtf(wred_sum(lv) * (1.0f / 256.0f) + 1e-5f);

  float o[8];
  #pragma unroll
  for (int i = 0; i < 8; ++i)
    o[i] = (v[i] - mu) * rstd * g[base + i] + bta[base + i];
  float4* o4 = (float4*)(OUT + outbase + base);
  o4[0] = make_float4(o[0], o[1], o[2], o[3]);
  o4[1] = make_float4(o[4], o[5], o[6], o[7]);
}

// ---------------------------------------------------------------------------
// Host orchestration
// ---------------------------------------------------------------------------
extern "C" void kernel_launch(void* const* d_in, const int* in_sizes, int n_in,
                              void* d_out, int out_size, void* d_ws, size_t ws_size,
                              hipStream_t stream) {
  (void)in_sizes; (void)n_in; (void)out_size; (void)ws_size;

  const float* q_emb = (const float*)d_in[0];
  const float* s_emb = (const float*)d_in[1];
  // d_in[2] = lens (unused by forward)
  auto P = [&](int i) { return (const float*)d_in[i]; };

  struct Blk { const float *Wo, *Wq, *Wv, *bo, *bq, *bv, *gam, *lnb, *lng; };
  // params pytree, sorted keys: b1,b2,b3,b4,know ; within: Wo,Wq,Wv,bo,bq,bv,gammas,ln_b,ln_g
  const Blk b1{P(3),  P(4),  P(5),  P(6),  P(7),  P(8),  P(9),  P(10), P(11)};
  const Blk b2{P(12), P(13), P(14), P(15), P(16), P(17), P(18), P(19), P(20)};
  const Blk b3{P(21), P(22), P(23), P(24), P(25), P(26), P(27), P(28), P(29)};
  // b4 sorted: Wk,Wo,Wq,Wv,bk,bo,bq,bv,gammas,ln_b,ln_g
  const float *Wk4 = P(30), *Wo4 = P(31), *Wq4 = P(32), *Wv4 = P(33);
  const float *bk4 = P(34), *bo4 = P(35), *bq4 = P(36), *bv4 = P(37);
  const float *gam4 = P(38), *lnb4 = P(39), *lng4 = P(40);
  const float *know = P(41);

  // workspace layout (floats)
  const size_t NTOK = (size_t)B_ * S_ * D_;          // 524288
  float* w      = (float*)d_ws;
  float* hq     = w;
  float* hs     = hq + NTOK;
  float* pp     = hs + NTOK;
  float* qkbuf  = pp + NTOK;
  float* vbuf   = qkbuf + NTOK;
  float* qb4    = vbuf + NTOK;                       // 4096 used
  float* rawb   = qb4 + 4096;                        // 8388608
  float* concat = rawb + 8388608;                    // 8388608 (b4 needs all)
  float* projo  = concat + 8388608;                  // 8388608

  float* out_z   = (float*)d_out;                    // [4,512,4096]
  float* out_qsc = out_z + (size_t)8388608;          // [4,8,512,512]
  float* out_ksc = out_qsc + (size_t)8388608;        // [4,8,512,16,512]

  const dim3 T32(32), T256(256);
  const dim3 Gp(128, 4);       // gemm over 2048 rows, 4 x 64-wide N panels
  const long XBSTR = (long)S_ * D_;

  auto std_block = [&](const float* qk_in, const float* v_in, const Blk& bp,
                       float* sc_out, float* h_out) {
    k_gemm_bias<<<Gp, T32, 0, stream>>>(qk_in, bp.Wq, bp.bq, qkbuf);   // q == k
    k_gemm_bias<<<Gp, T32, 0, stream>>>(v_in, bp.Wv, bp.bv, vbuf);
    k_qk<<<dim3(8, 32, 32), T32, 0, stream>>>(qkbuf, qkbuf, rawb, S_, XBSTR);
    k_rowwise<<<dim3(B_ * H_ * S_ / 8), T256, 0, stream>>>(rawb, sc_out, bp.gam,
                                                           /*peek=*/1, /*maxout=*/0, 0);
    k_av<<<dim3(32, 32), T32, 0, stream>>>(sc_out, vbuf, concat, 1);
    k_gemm_bias<<<Gp, T32, 0, stream>>>(concat, bp.Wo, bp.bo, projo);
    k_add_ln<<<dim3(B_ * S_ / 8), T256, 0, stream>>>(qk_in, projo, bp.lng, bp.lnb, h_out, 0);
  };

  // b1: self-attn on q_emb -> hq        (scores discarded -> in-place in rawb)
  std_block(q_emb, q_emb, b1, rawb, hq);
  // b2: self-attn on s_emb -> hs
  std_block(s_emb, s_emb, b2, rawb, hs);
  // b3: q=k from hq, v from hs -> p ; final scores are q_scores output
  std_block(hq, hs, b3, out_qsc, pp);

  // ---- b4: knowledge-expanded block (batch 64 via broadcast structure) ----
  k_gemm_bias<<<dim3(1, 4), T32, 0, stream>>>(know, Wq4, bq4, qb4);    // [16,256]
  k_gemm_bias<<<Gp, T32, 0, stream>>>(hq, Wk4, bk4, qkbuf);            // K proj
  k_gemm_bias<<<Gp, T32, 0, stream>>>(pp, Wv4, bv4, vbuf);             // V proj
  // raw scores, query-row independent: [ (b*H+h)*16 + j , ki ]
  k_qk<<<dim3(8, 1, 32), T32, 0, stream>>>(qb4, qkbuf, rawb, 16, 0L);
  // expand per query row, decay-softmax, maxout; write k_scores directly
  k_rowwise<<<dim3(B_ * H_ * NK_ * S_ / 8), T256, 0, stream>>>(rawb, out_ksc, gam4,
                                                               /*peek=*/0, /*maxout=*/1, 1);
  // A*V at expanded batch 64
  k_av<<<dim3(32, B_ * NK_ * H_), T32, 0, stream>>>(out_ksc, vbuf, concat, NK_);
  // output projection over 32768 rows
  k_gemm_bias<<<dim3(2048, 4), T32, 0, stream>>>(concat, Wo4, bo4, projo);
  // residual(know) + LN, scattered straight into z layout
  k_add_ln<<<dim3(B_ * NK_ * S_ / 8), T256, 0, stream>>>(know, projo, lng4, lnb4, out_z, 1);
}